// GATv2Layer_87393994539134
// MI455X (gfx1250) — compile-verified
//
#include <hip/hip_runtime.h>
#include <hip/hip_bf16.h>

typedef __attribute__((ext_vector_type(2))) float v2f;
typedef __attribute__((ext_vector_type(8))) float v8f;

#define D 128
#define H 4
#define NEG_SLOPE 0.2f
#define BN_EPS 1e-5f

// ---------- atomic helpers (lower to native CDNA5 global atomics) ----------
__device__ __forceinline__ void atomicMaxF(float* addr, float v) {
  // ordered-float trick: non-negative -> signed int max, negative -> unsigned min
  if (v >= 0.0f)
    atomicMax((int*)addr, __float_as_int(v));
  else
    atomicMin((unsigned int*)addr, __float_as_uint(v));
}

__device__ __forceinline__ void atomicAddF(float* addr, float v) {
  (void)__hip_atomic_fetch_add(addr, v, __ATOMIC_RELAXED, __HIP_MEMORY_SCOPE_AGENT);
}

// ---------- 0: workspace init (amax=-inf, accumulators=0) ----------
__global__ void init_ws_kernel(float* __restrict__ amax, float* __restrict__ denom,
                               float* __restrict__ aggout, float* __restrict__ colstats,
                               int NH, int ND) {
  int t = blockIdx.x * blockDim.x + threadIdx.x;
  if (t < NH) {
    amax[t]  = __int_as_float(0xff800000);  // -inf
    denom[t] = 0.0f;
  }
  if (t < ND) aggout[t] = 0.0f;
  if (t < 2 * D) colstats[t] = 0.0f;
}

// ---------- 1: dual projection GEMM via V_WMMA_F32_16X16X4_F32 ----------
// Block = 256 threads = 8 waves. Block computes a 16-row tile of xl AND xr.
// Wave w owns output columns [16w, 16w+16); per K-step it reuses one A
// fragment (staged through LDS) for two WMMAs (W_l and W_r).
__global__ __launch_bounds__(256) void proj_gemm_wmma(
    const float* __restrict__ hmat, const float* __restrict__ Wl,
    const float* __restrict__ Wr, const float* __restrict__ bl,
    const float* __restrict__ br, float* __restrict__ xl,
    float* __restrict__ xr, int N) {
  __shared__ float As[16 * 132];  // 16x128 A tile, stride 132 dodges bank conflicts
  const int row0 = blockIdx.x * 16;
  for (int i = threadIdx.x; i < 16 * D; i += 256) {
    int r = i >> 7, c = i & (D - 1);
    As[r * 132 + c] = (row0 + r < N) ? hmat[(size_t)(row0 + r) * D + c] : 0.0f;
  }
  __syncthreads();

  const int wave = threadIdx.x >> 5;       // n-tile 0..7
  const int lane = threadIdx.x & 31;
  const int m    = lane & 15;              // A row within tile
  const int koff = (lane >> 4) << 1;       // lanes 0-15: K={0,1}; 16-31: K={2,3}
  const int ncol = wave * 16 + m;          // B column this lane feeds

  v8f accL = {};
  v8f accR = {};
  for (int k = 0; k < D; k += 4) {
    v2f a, bL, bR;
    a[0]  = As[m * 132 + k + koff];
    a[1]  = As[m * 132 + k + koff + 1];
    bL[0] = Wl[(size_t)(k + koff) * D + ncol];
    bL[1] = Wl[(size_t)(k + koff + 1) * D + ncol];
    bR[0] = Wr[(size_t)(k + koff) * D + ncol];
    bR[1] = Wr[(size_t)(k + koff + 1) * D + ncol];
    accL = __builtin_amdgcn_wmma_f32_16x16x4_f32(false, a, false, bL, (short)0,
                                                 accL, false, false);
    accR = __builtin_amdgcn_wmma_f32_16x16x4_f32(false, a, false, bR, (short)0,
                                                 accR, false, false);
  }

  // C/D layout: VGPR i -> row (i + 8*(lane>=16)), col = wave*16 + lane%16
  const int ccol  = wave * 16 + (lane & 15);
  const int rbase = row0 + ((lane >> 4) << 3);
  const float bLc = bl[ccol];
  const float bRc = br[ccol];
#pragma unroll
  for (int i = 0; i < 8; ++i) {
    int r = rbase + i;
    if (r < N) {
      xl[(size_t)r * D + ccol] = accL[i] + bLc;
      xr[(size_t)r * D + ccol] = accR[i] + bRc;
    }
  }
}

// ---------- 2: per-edge GATv2 scores + segment max ----------
// One wave32 per edge; lane l covers channels [4l,4l+4); head = l/8.
__global__ __launch_bounds__(256) void edge_score_kernel(
    const float* __restrict__ xl, const float* __restrict__ xr,
    const int* __restrict__ ei, const float* __restrict__ att,
    float* __restrict__ alpha, float* __restrict__ amax, int E, int Et) {
  const int e = blockIdx.x * 8 + (threadIdx.x >> 5);
  if (e >= Et) return;
  const int lane = threadIdx.x & 31;
  int src, dst;
  if (e < E) { src = ei[e]; dst = ei[E + e]; }
  else       { src = e - E; dst = src; }          // self loop

  const float4 a = ((const float4*)(xl + (size_t)src * D))[lane];
  const float4 b = ((const float4*)(xr + (size_t)dst * D))[lane];
  const float4 w = ((const float4*)att)[lane];

  float x, s = 0.0f;
  x = a.x + b.x; s += w.x * (x >= 0.0f ? x : NEG_SLOPE * x);
  x = a.y + b.y; s += w.y * (x >= 0.0f ? x : NEG_SLOPE * x);
  x = a.z + b.z; s += w.z * (x >= 0.0f ? x : NEG_SLOPE * x);
  x = a.w + b.w; s += w.w * (x >= 0.0f ? x : NEG_SLOPE * x);

  // reduce within each 8-lane head group
  s += __shfl_xor(s, 1, 32);
  s += __shfl_xor(s, 2, 32);
  s += __shfl_xor(s, 4, 32);

  if ((lane & 7) == 0) {
    int hd = lane >> 3;
    alpha[(size_t)e * H + hd] = s;
    atomicMaxF(&amax[dst * H + hd], s);
  }
}

// ---------- 3: exp(alpha - max) + segment sum ----------
__global__ void edge_softmax_kernel(const int* __restrict__ ei,
                                    float* __restrict__ alpha,
                                    const float* __restrict__ amax,
                                    float* __restrict__ denom, int E, int Et) {
  int t = blockIdx.x * blockDim.x + threadIdx.x;
  if (t >= Et * H) return;
  int e = t >> 2, hd = t & 3;
  int dst = (e < E) ? ei[E + e] : (e - E);
  float v = __expf(alpha[t] - amax[dst * H + hd]);
  alpha[t] = v;
  atomicAddF(&denom[dst * H + hd], v);
}

// ---------- 4: normalized scatter aggregation ----------
__global__ __launch_bounds__(256) void edge_agg_kernel(
    const float* __restrict__ xl, const int* __restrict__ ei,
    const float* __restrict__ alpha, const float* __restrict__ denom,
    float* __restrict__ aggout, int E, int Et) {
  const int e = blockIdx.x * 8 + (threadIdx.x >> 5);
  if (e >= Et) return;
  const int lane = threadIdx.x & 31;
  int src, dst;
  if (e < E) { src = ei[e]; dst = ei[E + e]; }
  else       { src = e - E; dst = src; }
  const int hd  = lane >> 3;
  const float w = alpha[(size_t)e * H + hd] / (denom[dst * H + hd] + 1e-16f);
  const float4 v = ((const float4*)(xl + (size_t)src * D))[lane];
  float* o = aggout + (size_t)dst * D + lane * 4;
  atomicAddF(o + 0, w * v.x);
  atomicAddF(o + 1, w * v.y);
  atomicAddF(o + 2, w * v.z);
  atomicAddF(o + 3, w * v.w);
}

// ---------- 5: BN column statistics (sum, sum of squares) ----------
__global__ __launch_bounds__(256) void bn_stats_kernel(
    const float* __restrict__ aggout, float* __restrict__ colstats, int N) {
  const int d = threadIdx.x & (D - 1);
  const int rpb = blockDim.x >> 7;  // rows per block per iteration (2)
  int r = blockIdx.x * rpb + (threadIdx.x >> 7);
  const int rstride = gridDim.x * rpb;
  float s = 0.0f, s2 = 0.0f;
  for (; r < N; r += rstride) {
    float v = aggout[(size_t)r * D + d];
    s += v; s2 += v * v;
  }
  atomicAddF(&colstats[d], s);
  atomicAddF(&colstats[D + d], s2);
}

// ---------- 6: BN normalize + ReLU + residual ----------
// bias_out cancels exactly inside BatchNorm (shifts mean identically), so skip it.
__global__ void bn_final_kernel(const float* __restrict__ aggout,
                                const float* __restrict__ colstats,
                                const float* __restrict__ gamma,
                                const float* __restrict__ beta,
                                const float* __restrict__ hmat,
                                float* __restrict__ out, int N) {
  int t = blockIdx.x * blockDim.x + threadIdx.x;
  if (t >= N * D) return;
  int d = t & (D - 1);
  float invN = 1.0f / (float)N;
  float mu   = colstats[d] * invN;
  float var  = colstats[D + d] * invN - mu * mu;   // E[x^2] - mu^2
  float v = (aggout[t] - mu) * rsqrtf(var + BN_EPS) * gamma[d] + beta[d];
  v = v > 0.0f ? v : 0.0f;
  out[t] = v + hmat[t];
}

extern "C" void kernel_launch(void* const* d_in, const int* in_sizes, int n_in,
                              void* d_out, int out_size, void* d_ws, size_t ws_size,
                              hipStream_t stream) {
  const float* hmat  = (const float*)d_in[0];
  const int*   ei    = (const int*)d_in[1];
  const float* Wl    = (const float*)d_in[2];
  const float* Wr    = (const float*)d_in[3];
  const float* bl    = (const float*)d_in[4];
  const float* br    = (const float*)d_in[5];
  const float* att   = (const float*)d_in[6];
  /* d_in[7] = bias_out: mathematically cancels through BatchNorm */
  const float* gamma = (const float*)d_in[8];
  const float* beta  = (const float*)d_in[9];

  const int N  = in_sizes[0] / D;
  const int E  = in_sizes[1] / 2;
  const int Et = E + N;  // with self loops

  // workspace layout (floats): xl | xr | alpha | amax | denom | aggout | colstats
  float* ws       = (float*)d_ws;
  float* xl       = ws;
  float* xr       = xl     + (size_t)N * D;
  float* alpha    = xr     + (size_t)N * D;
  float* amax     = alpha  + (size_t)Et * H;
  float* denom    = amax   + (size_t)N * H;
  float* aggout   = denom  + (size_t)N * H;
  float* colstats = aggout + (size_t)N * D;

  const int ND = N * D;
  const int NH = N * H;

  init_ws_kernel<<<(ND + 255) / 256, 256, 0, stream>>>(amax, denom, aggout, colstats, NH, ND);
  proj_gemm_wmma<<<(N + 15) / 16, 256, 0, stream>>>(hmat, Wl, Wr, bl, br, xl, xr, N);
  edge_score_kernel<<<(Et + 7) / 8, 256, 0, stream>>>(xl, xr, ei, att, alpha, amax, E, Et);
  edge_softmax_kernel<<<(Et * H + 255) / 256, 256, 0, stream>>>(ei, alpha, amax, denom, E, Et);
  edge_agg_kernel<<<(Et + 7) / 8, 256, 0, stream>>>(xl, ei, alpha, denom, aggout, E, Et);
  bn_stats_kernel<<<512, 256, 0, stream>>>(aggout, colstats, N);
  bn_final_kernel<<<(ND + 255) / 256, 256, 0, stream>>>(aggout, colstats, gamma, beta, hmat,
                                                        (float*)d_out, N);
}